// AttnDecoderRNN_58222576665170
// MI455X (gfx1250) — compile-verified
//
#include <hip/hip_runtime.h>
#include <hip/hip_bf16.h>

// CDNA5 / gfx1250, wave32. fp32 WMMA (V_WMMA_F32_16X16X4_F32) for all GEMMs.
typedef __attribute__((ext_vector_type(2))) float v2f;
typedef __attribute__((ext_vector_type(8))) float v8f;

constexpr int kB = 128;
constexpr int kH = 1024;
constexpr int kL = 64;
constexpr int kV = 32000;

// ---------------------------------------------------------------------------
// Generic C[M,N] = A[M,K] @ W[N,K]^T + bias[N].  One wave per 16x16 C tile.
// A fragment: lane&15 = M row, half=lane>>4 selects K pair {0,1}/{2,3}.
// B fragment mirrors A with lane&15 = N column (W row).
// ---------------------------------------------------------------------------
__global__ __launch_bounds__(32) void gemm_xwt_bias(
    const float* __restrict__ A, const float* __restrict__ W,
    const float* __restrict__ bias, float* __restrict__ C,
    int N, int K) {
  const int lane = threadIdx.x;
  const int half = lane >> 4, idx = lane & 15;
  const long n0 = (long)blockIdx.x * 16, m0 = (long)blockIdx.y * 16;
  const float* __restrict__ arow = A + (m0 + idx) * (long)K;
  const float* __restrict__ wrow = W + (n0 + idx) * (long)K;
  v8f acc = {};
  for (int k = 0; k < K; k += 4) {
    v2f a, b;
    a.x = arow[k + 2 * half]; a.y = arow[k + 2 * half + 1];
    b.x = wrow[k + 2 * half]; b.y = wrow[k + 2 * half + 1];
    acc = __builtin_amdgcn_wmma_f32_16x16x4_f32(false, a, false, b,
                                                (short)0, acc, false, false);
  }
  const float bn = bias[n0 + idx];
  for (int r = 0; r < 8; ++r)
    C[(m0 + half * 8 + r) * (long)N + n0 + idx] = acc[r] + bn;
}

// ---------------------------------------------------------------------------
// gates = x @ W_ih^T + b_ih + h0 @ W_hh^T + b_hh   (two K loops, one fragment)
// ---------------------------------------------------------------------------
__global__ __launch_bounds__(32) void gemm_gates(
    const float* __restrict__ X, const float* __restrict__ Wih,
    const float* __restrict__ h0, const float* __restrict__ Whh,
    const float* __restrict__ bih, const float* __restrict__ bhh,
    float* __restrict__ gates) {
  const int lane = threadIdx.x;
  const int half = lane >> 4, idx = lane & 15;
  const long n0 = (long)blockIdx.x * 16, m0 = (long)blockIdx.y * 16;
  const float* __restrict__ ax = X   + (m0 + idx) * (long)(3 * kH);
  const float* __restrict__ wx = Wih + (n0 + idx) * (long)(3 * kH);
  const float* __restrict__ ah = h0  + (m0 + idx) * (long)kH;
  const float* __restrict__ wh = Whh + (n0 + idx) * (long)kH;
  v8f acc = {};
  for (int k = 0; k < 3 * kH; k += 4) {
    v2f a, b;
    a.x = ax[k + 2 * half]; a.y = ax[k + 2 * half + 1];
    b.x = wx[k + 2 * half]; b.y = wx[k + 2 * half + 1];
    acc = __builtin_amdgcn_wmma_f32_16x16x4_f32(false, a, false, b,
                                                (short)0, acc, false, false);
  }
  for (int k = 0; k < kH; k += 4) {
    v2f a, b;
    a.x = ah[k + 2 * half]; a.y = ah[k + 2 * half + 1];
    b.x = wh[k + 2 * half]; b.y = wh[k + 2 * half + 1];
    acc = __builtin_amdgcn_wmma_f32_16x16x4_f32(false, a, false, b,
                                                (short)0, acc, false, false);
  }
  const float bn = bih[n0 + idx] + bhh[n0 + idx];
  for (int r = 0; r < 8; ++r)
    gates[(m0 + half * 8 + r) * (long)(4 * kH) + n0 + idx] = acc[r] + bn;
}

// ---------------------------------------------------------------------------
// Fused attention scores: for rows m = b*L+l, tile over n (H):
//   partial[m][ntile] = sum_{n in tile} v[n]*tanh(wa_s[b,n] + bUa[n] + (Ua@enc_m)[n])
// Avoids materializing the 33.5 MB ua_e tensor.
// ---------------------------------------------------------------------------
__global__ __launch_bounds__(32) void attn_score_partials(
    const float* __restrict__ enc,   // [B*L, 2H] row-major
    const float* __restrict__ Ua,    // [H, 2H]
    const float* __restrict__ wa_s,  // [B, H]
    const float* __restrict__ bUa,   // [H]
    const float* __restrict__ v,     // [H]
    float* __restrict__ partials) {  // [B*L, 64]
  const int lane = threadIdx.x;
  const int half = lane >> 4, idx = lane & 15;
  const int ntile = blockIdx.x;                       // 0..63
  const long n0 = (long)ntile * 16, m0 = (long)blockIdx.y * 16;
  const int b = (int)(m0 >> 6);                       // 16-row tile stays in one b
  const float* __restrict__ arow = enc + (m0 + idx) * (long)(2 * kH);
  const float* __restrict__ wrow = Ua  + (n0 + idx) * (long)(2 * kH);
  v8f acc = {};
  for (int k = 0; k < 2 * kH; k += 4) {
    v2f a, bb;
    a.x  = arow[k + 2 * half]; a.y  = arow[k + 2 * half + 1];
    bb.x = wrow[k + 2 * half]; bb.y = wrow[k + 2 * half + 1];
    acc = __builtin_amdgcn_wmma_f32_16x16x4_f32(false, a, false, bb,
                                                (short)0, acc, false, false);
  }
  const float wsn = wa_s[(long)b * kH + n0 + idx] + bUa[n0 + idx];
  const float vn  = v[n0 + idx];
  for (int r = 0; r < 8; ++r) {
    float p = vn * tanhf(wsn + acc[r]);
    p += __shfl_xor(p, 1, 16);
    p += __shfl_xor(p, 2, 16);
    p += __shfl_xor(p, 4, 16);
    p += __shfl_xor(p, 8, 16);
    if (idx == 0)
      partials[(m0 + half * 8 + r) * 64 + ntile] = p;
  }
}

// ---------------------------------------------------------------------------
// Deterministic reduction of partials + mask + softmax over L=64.
// ---------------------------------------------------------------------------
__global__ __launch_bounds__(kL) void attn_softmax(
    const float* __restrict__ partials, const int* __restrict__ mask,
    float* __restrict__ attn) {
  __shared__ float red[kL];
  const int b = blockIdx.x, l = threadIdx.x;
  const float* __restrict__ p = partials + ((long)b * kL + l) * 64;
  float sc = 0.f;
  for (int t = 0; t < 64; ++t) sc += p[t];       // fixed order -> deterministic
  if (mask[b * kL + l] == 0) sc = -1e9f;
  red[l] = sc; __syncthreads();
  for (int o = 32; o > 0; o >>= 1) { if (l < o) red[l] = fmaxf(red[l], red[l + o]); __syncthreads(); }
  const float mx = red[0]; __syncthreads();
  const float e = __expf(sc - mx);
  red[l] = e; __syncthreads();
  for (int o = 32; o > 0; o >>= 1) { if (l < o) red[l] += red[l + o]; __syncthreads(); }
  attn[(long)b * kL + l] = e / red[0];
}

// ---------------------------------------------------------------------------
// Gather embedding rows into x[:, 0:H]
// ---------------------------------------------------------------------------
__global__ __launch_bounds__(256) void embed_kernel(
    const int* __restrict__ ids, const float* __restrict__ emb,
    float* __restrict__ x) {
  const int b = blockIdx.y;
  const int j = blockIdx.x * 256 + threadIdx.x;
  x[(long)b * (3 * kH) + j] = emb[(long)ids[b] * kH + j];
}

// ---------------------------------------------------------------------------
// ctx[b,e] = sum_l attn[b,l] * enc[b,l,e]  -> x[:, H:3H]
// ---------------------------------------------------------------------------
__global__ __launch_bounds__(256) void ctx_kernel(
    const float* __restrict__ attn, const float* __restrict__ enc,
    float* __restrict__ x) {
  const int b = blockIdx.y;
  const int e = blockIdx.x * 256 + threadIdx.x;      // e < 2H
  const float* __restrict__ ar = attn + (long)b * kL;
  const float* __restrict__ er = enc + (long)b * kL * (2 * kH) + e;
  float acc = 0.f;
  for (int l = 0; l < kL; ++l) acc += ar[l] * er[(long)l * (2 * kH)];
  x[(long)b * (3 * kH) + kH + e] = acc;
}

// ---------------------------------------------------------------------------
// LSTM cell elementwise (PyTorch gate order i,f,g,o)
// ---------------------------------------------------------------------------
__global__ __launch_bounds__(256) void lstm_kernel(
    const float* __restrict__ gates, const float* __restrict__ c0,
    float* __restrict__ h1, float* __restrict__ c1) {
  const long t = (long)blockIdx.x * 256 + threadIdx.x;   // < B*H
  const int b = (int)(t / kH), h = (int)(t % kH);
  const float* __restrict__ g = gates + (long)b * (4 * kH);
  const float ig = 1.f / (1.f + __expf(-g[h]));
  const float fg = 1.f / (1.f + __expf(-g[kH + h]));
  const float gg = tanhf(g[2 * kH + h]);
  const float og = 1.f / (1.f + __expf(-g[3 * kH + h]));
  const float c = fg * c0[t] + ig * gg;
  c1[t] = c;
  h1[t] = og * tanhf(c);
}

// ---------------------------------------------------------------------------
// In-place log_softmax over V=32000 per row.
// ---------------------------------------------------------------------------
__global__ __launch_bounds__(256) void logsoftmax_kernel(float* __restrict__ lp) {
  __shared__ float red[256];
  const int b = blockIdx.x, t = threadIdx.x;
  float* __restrict__ row = lp + (long)b * kV;
  float mx = -3.402823466e38f;
  for (int j = t; j < kV; j += 256) mx = fmaxf(mx, row[j]);
  red[t] = mx; __syncthreads();
  for (int o = 128; o > 0; o >>= 1) { if (t < o) red[t] = fmaxf(red[t], red[t + o]); __syncthreads(); }
  mx = red[0]; __syncthreads();
  float s = 0.f;
  for (int j = t; j < kV; j += 256) s += __expf(row[j] - mx);
  red[t] = s; __syncthreads();
  for (int o = 128; o > 0; o >>= 1) { if (t < o) red[t] += red[t + o]; __syncthreads(); }
  const float lse = mx + __logf(red[0]);
  for (int j = t; j < kV; j += 256) row[j] -= lse;
}

extern "C" void kernel_launch(void* const* d_in, const int* in_sizes, int n_in,
                              void* d_out, int out_size, void* d_ws, size_t ws_size,
                              hipStream_t stream) {
  (void)in_sizes; (void)n_in; (void)out_size; (void)ws_size;
  const int*   ids  = (const int*)  d_in[0];
  const float* h0   = (const float*)d_in[1];
  const float* c0   = (const float*)d_in[2];
  const float* enc  = (const float*)d_in[3];
  const int*   mask = (const int*)  d_in[4];
  const float* emb  = (const float*)d_in[5];
  const float* Wa   = (const float*)d_in[6];
  const float* ba   = (const float*)d_in[7];
  const float* Ua   = (const float*)d_in[8];
  const float* bUa  = (const float*)d_in[9];
  const float* v    = (const float*)d_in[10];
  const float* Wih  = (const float*)d_in[11];
  const float* Whh  = (const float*)d_in[12];
  const float* bih  = (const float*)d_in[13];
  const float* bhh  = (const float*)d_in[14];
  const float* Wout = (const float*)d_in[15];
  const float* bout = (const float*)d_in[16];

  float* out       = (float*)d_out;          // outputs concatenated flat
  float* log_probs = out;                    // [B,V]
  float* h1        = log_probs + (long)kB * kV;   // [B,H]
  float* c1        = h1 + (long)kB * kH;          // [B,H]
  float* attn      = c1 + (long)kB * kH;          // [B,L]

  float* ws       = (float*)d_ws;            // all written before read
  float* wa_s     = ws;                            // B*H
  float* x        = wa_s + (long)kB * kH;          // B*3H  ([embedded | ctx])
  float* gates    = x + (long)kB * 3 * kH;         // B*4H
  float* partials = gates + (long)kB * 4 * kH;     // B*L*64

  // 1) embedding gather -> x[:, :H]
  embed_kernel<<<dim3(kH / 256, kB), 256, 0, stream>>>(ids, emb, x);
  // 2) wa_s = h0 @ Wa^T + ba                       (WMMA f32)
  gemm_xwt_bias<<<dim3(kH / 16, kB / 16), 32, 0, stream>>>(h0, Wa, ba, wa_s, kH, kH);
  // 3) fused attention score partials              (WMMA f32, 34 GF GEMM)
  attn_score_partials<<<dim3(kH / 16, (kB * kL) / 16), 32, 0, stream>>>(
      enc, Ua, wa_s, bUa, v, partials);
  // 4) mask + softmax -> attn weights (d_out)
  attn_softmax<<<dim3(kB), kL, 0, stream>>>(partials, mask, attn);
  // 5) ctx -> x[:, H:3H]
  ctx_kernel<<<dim3((2 * kH) / 256, kB), 256, 0, stream>>>(attn, enc, x);
  // 6) gates = x@Wih^T + bih + h0@Whh^T + bhh      (WMMA f32)
  gemm_gates<<<dim3((4 * kH) / 16, kB / 16), 32, 0, stream>>>(
      x, Wih, h0, Whh, bih, bhh, gates);
  // 7) LSTM cell -> h1, c1 (d_out)
  lstm_kernel<<<dim3((kB * kH) / 256), 256, 0, stream>>>(gates, c0, h1, c1);
  // 8) logits = h1 @ Wout^T + bout -> d_out        (WMMA f32, 131 MB stream)
  gemm_xwt_bias<<<dim3(kV / 16, kB / 16), 32, 0, stream>>>(h1, Wout, bout, log_probs, kV, kH);
  // 9) in-place log_softmax
  logsoftmax_kernel<<<dim3(kB), 256, 0, stream>>>(log_probs);
}